// ImprovedLSTMRegressor_63660005262026
// MI455X (gfx1250) — compile-verified
//
#include <hip/hip_runtime.h>
#include <math.h>
#include <stddef.h>

// ---------------------------------------------------------------------------
// ImprovedLSTMRegressor for MI455X (gfx1250), fp32 end-to-end.
// Matrix work uses V_WMMA_F32_16X16X4_F32 (exact fp32 semantics).
//
//   K1 conv_blocks : fused residual conv blocks, register-blocked VALU
//   K2 wmma_gemm   : double-buffered fp32 WMMA GEMM (input projections, attn)
//   K3 lstm_layer  : recurrence; Whh in LDS, gate rows prefetched, WMMA
//   K4 attn_heads  : softmax pooling + MLP heads
// ---------------------------------------------------------------------------

#define B_     64
#define T_     256
#define P_     21
#define D_     3
#define C1_    32
#define C2_    64
#define FEAT_  1344          // C2_*P_
#define G4_    512           // 2 dirs * 4*HID
#define NROWS_ 16384         // B_*T_
#define EPS_   1e-5f
#define ROWP_  26            // padded conv row: col0 pad, cols 22..25 pad

typedef __attribute__((ext_vector_type(2))) float v2f;
typedef __attribute__((ext_vector_type(4))) float v4f;
typedef __attribute__((ext_vector_type(8))) float v8f;

// fp32 WMMA: D(16x16) = A(16x4) * B(4x16) + C
__device__ __forceinline__ v8f wmma_f32(v2f a, v2f b, v8f c) {
  return __builtin_amdgcn_wmma_f32_16x16x4_f32(false, a, false, b, (short)0, c,
                                               false, false);
}

// Fragment loader (A 16x4, or W^T-as-B stored [n][k]):
// lane L<16 -> {M[L][kk], M[L][kk+1]}, lane L>=16 -> {M[L-16][kk+2], ...+3}.
__device__ __forceinline__ v2f ld_frag(const float* p, int row0, int ldk,
                                       int kk, int lane) {
  int r = row0 + (lane & 15);
  int k = kk + ((lane >> 4) << 1);
  return *(const v2f*)(p + (size_t)r * ldk + k);
}

// ---------------------------------------------------------------------------
// K1: fused residual conv blocks. 4 frames / 256-thread block, 64 thr/frame.
// Heavy k=3 stages register-blocked: thread = (channel group) x (pos group).
// ---------------------------------------------------------------------------
struct ConvParams {
  const float *c1w, *c1b, *bn1g, *bn1b, *bn1m, *bn1v;
  const float *c2w, *c2b, *bn2g, *bn2b, *bn2m, *bn2v;
  const float *scw, *scb, *sbg, *sbb, *sbm, *sbv;
};

__device__ __forceinline__ float bn_apply(float v, const float* g,
                                          const float* b, const float* m,
                                          const float* var, int c) {
  return (v - m[c]) * (g[c] * rsqrtf(var[c] + EPS_)) + b[c];
}

__global__ __launch_bounds__(256) void conv_blocks(
    const float* __restrict__ x, ConvParams p1, ConvParams p2,
    float* __restrict__ conv_seq) {
  __shared__ float xf[4][3][ROWP_];
  __shared__ float t1[4][32][ROWP_];
  __shared__ float y1[4][32][ROWP_];
  __shared__ float t3[4][64][ROWP_];

  const int tid = threadIdx.x;
  const int f = tid >> 6;           // frame slot within block
  const int lt = tid & 63;          // 64 threads per frame
  const int n = blockIdx.x * 4 + f; // global frame = b*T_ + t
  const int cg = lt & 15;           // channel group
  const int pg = lt >> 4;           // position group (4)
  const int p0 = pg * 6;
  const int np = (21 - p0) < 6 ? (21 - p0) : 6;  // 6,6,6,3

  // zero pads
  for (int i = lt; i < 3 * ROWP_; i += 64) xf[f][i / ROWP_][i % ROWP_] = 0.f;
  for (int i = lt; i < 32; i += 64) {
    t1[f][i][0] = 0.f; y1[f][i][0] = 0.f;
#pragma unroll
    for (int cpd = 22; cpd < ROWP_; cpd++) {
      t1[f][i][cpd] = 0.f; y1[f][i][cpd] = 0.f;
    }
  }
  for (int i = lt; i < 64; i += 64) {
    t3[f][i][0] = 0.f;
#pragma unroll
    for (int cpd = 22; cpd < ROWP_; cpd++) t3[f][i][cpd] = 0.f;
  }
  __syncthreads();

  // load frame: frames[n,d,p] = x[n*63 + p*3 + d]
  for (int i = lt; i < P_ * D_; i += 64) {
    int pp = i / 3, d = i % 3;
    xf[f][d][1 + pp] = x[(size_t)n * 63 + i];
  }
  __syncthreads();

  // ---- block1 conv1 (32<-3,k3) + bn1 + relu (tiny K, per-output loop) ----
  for (int idx = lt; idx < C1_ * P_; idx += 64) {
    int c = idx / P_, pp = idx % P_;
    float s = p1.c1b[c];
#pragma unroll
    for (int ci = 0; ci < 3; ci++)
#pragma unroll
      for (int kw = 0; kw < 3; kw++)
        s += p1.c1w[c * 9 + ci * 3 + kw] * xf[f][ci][pp + kw];
    s = bn_apply(s, p1.bn1g, p1.bn1b, p1.bn1m, p1.bn1v, c);
    t1[f][c][1 + pp] = fmaxf(s, 0.f);
  }
  __syncthreads();

  // ---- block1 conv2 (32<-32,k3)+bn2, shortcut 1x1+bn, relu: CB=2 ----
  {
    const int c0 = cg * 2;
    float acc[2][6];
#pragma unroll
    for (int u = 0; u < 2; u++)
#pragma unroll
      for (int j = 0; j < 6; j++) acc[u][j] = 0.f;
    for (int ci = 0; ci < 32; ci++) {
      float in[8];
#pragma unroll
      for (int j = 0; j < 8; j++) in[j] = t1[f][ci][p0 + j];
      float wv[2][3];
#pragma unroll
      for (int u = 0; u < 2; u++)
#pragma unroll
        for (int kw = 0; kw < 3; kw++)
          wv[u][kw] = p1.c2w[(c0 + u) * 96 + ci * 3 + kw];
#pragma unroll
      for (int u = 0; u < 2; u++)
#pragma unroll
        for (int j = 0; j < 6; j++)
          acc[u][j] += wv[u][0] * in[j] + wv[u][1] * in[j + 1] +
                       wv[u][2] * in[j + 2];
    }
#pragma unroll
    for (int u = 0; u < 2; u++) {
      int c = c0 + u;
      for (int j = 0; j < np; j++) {
        int pp = p0 + j;
        float s = acc[u][j] + p1.c2b[c];
        s = bn_apply(s, p1.bn2g, p1.bn2b, p1.bn2m, p1.bn2v, c);
        float sc = p1.scb[c];
#pragma unroll
        for (int ci = 0; ci < 3; ci++)
          sc += p1.scw[c * 3 + ci] * xf[f][ci][1 + pp];
        sc = bn_apply(sc, p1.sbg, p1.sbb, p1.sbm, p1.sbv, c);
        y1[f][c][1 + pp] = fmaxf(s + sc, 0.f);
      }
    }
  }
  __syncthreads();

  // ---- block2 conv1 (64<-32,k3) + bn1 + relu: CB=4 ----
  {
    const int c0 = cg * 4;
    float acc[4][6];
#pragma unroll
    for (int u = 0; u < 4; u++)
#pragma unroll
      for (int j = 0; j < 6; j++) acc[u][j] = 0.f;
    for (int ci = 0; ci < 32; ci++) {
      float in[8];
#pragma unroll
      for (int j = 0; j < 8; j++) in[j] = y1[f][ci][p0 + j];
      float wv[4][3];
#pragma unroll
      for (int u = 0; u < 4; u++)
#pragma unroll
        for (int kw = 0; kw < 3; kw++)
          wv[u][kw] = p2.c1w[(c0 + u) * 96 + ci * 3 + kw];
#pragma unroll
      for (int u = 0; u < 4; u++)
#pragma unroll
        for (int j = 0; j < 6; j++)
          acc[u][j] += wv[u][0] * in[j] + wv[u][1] * in[j + 1] +
                       wv[u][2] * in[j + 2];
    }
#pragma unroll
    for (int u = 0; u < 4; u++) {
      int c = c0 + u;
      for (int j = 0; j < np; j++) {
        float s = acc[u][j] + p2.c1b[c];
        s = bn_apply(s, p2.bn1g, p2.bn1b, p2.bn1m, p2.bn1v, c);
        t3[f][c][1 + p0 + j] = fmaxf(s, 0.f);
      }
    }
  }
  __syncthreads();

  // ---- block2 conv2 (64<-64,k3)+bn2, shortcut(64<-32)+bn, relu: CB=4 ----
  {
    const int c0 = cg * 4;
    float acc[4][6];
#pragma unroll
    for (int u = 0; u < 4; u++)
#pragma unroll
      for (int j = 0; j < 6; j++) acc[u][j] = 0.f;
    for (int ci = 0; ci < 64; ci++) {
      float in[8];
#pragma unroll
      for (int j = 0; j < 8; j++) in[j] = t3[f][ci][p0 + j];
      float wv[4][3];
#pragma unroll
      for (int u = 0; u < 4; u++)
#pragma unroll
        for (int kw = 0; kw < 3; kw++)
          wv[u][kw] = p2.c2w[(c0 + u) * 192 + ci * 3 + kw];
#pragma unroll
      for (int u = 0; u < 4; u++)
#pragma unroll
        for (int j = 0; j < 6; j++)
          acc[u][j] += wv[u][0] * in[j] + wv[u][1] * in[j + 1] +
                       wv[u][2] * in[j + 2];
    }
#pragma unroll
    for (int u = 0; u < 4; u++) {
      int c = c0 + u;
      for (int j = 0; j < np; j++) {
        int pp = p0 + j;
        float s = acc[u][j] + p2.c2b[c];
        s = bn_apply(s, p2.bn2g, p2.bn2b, p2.bn2m, p2.bn2v, c);
        float sc = p2.scb[c];
        for (int ci = 0; ci < 32; ci++)
          sc += p2.scw[c * 32 + ci] * y1[f][ci][1 + pp];
        sc = bn_apply(sc, p2.sbg, p2.sbb, p2.sbm, p2.sbv, c);
        conv_seq[(size_t)n * FEAT_ + c * P_ + pp] = fmaxf(s + sc, 0.f);
      }
    }
  }
}

// ---------------------------------------------------------------------------
// K2: double-buffered fp32 WMMA GEMM  Out[M,N] = X[M,K] @ W[N,K]^T + bias,
// optional tanh. W split at nsplit rows across two pointers.
// Block tile 128x64, K chunk 32, 256 threads = 8 waves, 4 accs/wave.
// Requires: M%128==0, N%64==0, K%32==0.
// ---------------------------------------------------------------------------
__global__ __launch_bounds__(256) void wmma_gemm_xwT(
    const float* __restrict__ X, const float* __restrict__ W0,
    const float* __restrict__ W1, int nsplit, const float* ba0,
    const float* bb0, const float* ba1, const float* bb1,
    float* __restrict__ Out, int M, int N, int K, int act) {
  __shared__ float As[2][128 * 32];
  __shared__ float Bs[2][64 * 32];
  const int tid = threadIdx.x;
  const int lane = tid & 31;
  const int wv = tid >> 5;
  const int m0 = blockIdx.y * 128;
  const int n0 = blockIdx.x * 64;

  v8f acc[4] = {};
  v4f ra[4], rb[2];

  // preload chunk 0
#pragma unroll
  for (int i = 0; i < 4; i++) {
    int s = tid + i * 256, r = s >> 3, kv = (s & 7) << 2;
    ra[i] = *(const v4f*)&X[(size_t)(m0 + r) * K + kv];
  }
#pragma unroll
  for (int i = 0; i < 2; i++) {
    int s = tid + i * 256, r = s >> 3, kv = (s & 7) << 2;
    int j = n0 + r;
    const float* wrow =
        (j < nsplit) ? (W0 + (size_t)j * K) : (W1 + (size_t)(j - nsplit) * K);
    rb[i] = *(const v4f*)&wrow[kv];
  }
#pragma unroll
  for (int i = 0; i < 4; i++) {
    int s = tid + i * 256, r = s >> 3, kv = (s & 7) << 2;
    *(v4f*)&As[0][r * 32 + kv] = ra[i];
  }
#pragma unroll
  for (int i = 0; i < 2; i++) {
    int s = tid + i * 256, r = s >> 3, kv = (s & 7) << 2;
    *(v4f*)&Bs[0][r * 32 + kv] = rb[i];
  }
  __syncthreads();

  int cur = 0;
  for (int kk0 = 0; kk0 < K; kk0 += 32) {
    const bool more = (kk0 + 32) < K;
    if (more) {  // stage next chunk into registers (overlaps WMMA below)
#pragma unroll
      for (int i = 0; i < 4; i++) {
        int s = tid + i * 256, r = s >> 3, kv = (s & 7) << 2;
        ra[i] = *(const v4f*)&X[(size_t)(m0 + r) * K + kk0 + 32 + kv];
      }
#pragma unroll
      for (int i = 0; i < 2; i++) {
        int s = tid + i * 256, r = s >> 3, kv = (s & 7) << 2;
        int j = n0 + r;
        const float* wrow = (j < nsplit) ? (W0 + (size_t)j * K)
                                         : (W1 + (size_t)(j - nsplit) * K);
        rb[i] = *(const v4f*)&wrow[kk0 + 32 + kv];
      }
      if (kk0 + 64 < K) {  // global_prefetch_b8 for chunk after next
        int r = tid >> 3, kv = (tid & 7) << 2;
        __builtin_prefetch(&X[(size_t)(m0 + r) * K + kk0 + 64 + kv], 0, 0);
      }
    }
    const float* Ac = As[cur];
    const float* Bc = Bs[cur];
#pragma unroll
    for (int ks = 0; ks < 8; ks++) {
      int kk = ks * 4;
      v2f a = ld_frag(Ac, wv * 16, 32, kk, lane);
#pragma unroll
      for (int nt = 0; nt < 4; nt++) {
        v2f b = ld_frag(Bc, nt * 16, 32, kk, lane);
        acc[nt] = wmma_f32(a, b, acc[nt]);
      }
    }
    if (more) {
      const int nxt = cur ^ 1;
#pragma unroll
      for (int i = 0; i < 4; i++) {
        int s = tid + i * 256, r = s >> 3, kv = (s & 7) << 2;
        *(v4f*)&As[nxt][r * 32 + kv] = ra[i];
      }
#pragma unroll
      for (int i = 0; i < 2; i++) {
        int s = tid + i * 256, r = s >> 3, kv = (s & 7) << 2;
        *(v4f*)&Bs[nxt][r * 32 + kv] = rb[i];
      }
    }
    __syncthreads();
    cur ^= 1;
  }

  // epilogue
  const int rbase = m0 + wv * 16 + ((lane >> 4) << 3);
  const int cl = lane & 15;
#pragma unroll
  for (int nt = 0; nt < 4; nt++) {
    int col = n0 + nt * 16 + cl;
    float bias = (col < nsplit)
                     ? (ba0[col] + (bb0 ? bb0[col] : 0.f))
                     : (ba1[col - nsplit] + (bb1 ? bb1[col - nsplit] : 0.f));
#pragma unroll
    for (int i = 0; i < 8; i++) {
      float v = acc[nt][i] + bias;
      if (act == 1) v = tanhf(v);
      Out[(size_t)(rbase + i) * N + col] = v;
    }
  }
}

// ---------------------------------------------------------------------------
// K3: one BiLSTM layer recurrence. Grid = 8 blocks: dir = bid>>2, batch
// quarter = bid&3 (16 rows). Whh staged in LDS; gate rows prefetched into
// registers before the WMMA chain; split accumulators halve the RAW chain.
// ---------------------------------------------------------------------------
__global__ __launch_bounds__(256) void lstm_layer(
    const float* __restrict__ G, const float* __restrict__ WhhF,
    const float* __restrict__ WhhB, const int* __restrict__ lengths,
    float* __restrict__ Hout, int maskOutput) {
  __shared__ float whh_s[256 * 64];   // 64 KB (gfx1250: up to 320KB LDS/WG)
  __shared__ float h_s[16 * 64];
  __shared__ float c_s[16 * 64];
  __shared__ float g_s[16 * 256];
  __shared__ int len_s[16];

  const int tid = threadIdx.x;
  const int lane = tid & 31;
  const int wv = tid >> 5;
  const int dir = blockIdx.x >> 2;
  const int b0 = (blockIdx.x & 3) * 16;
  const float* Whh = dir ? WhhB : WhhF;   // [256 gates][64 k] = B^T layout
  const int dirOff = dir * 256;
  const int outOff = dir * 64;

  for (int i = tid; i < 4096; i += 256)
    *(v4f*)&whh_s[i * 4] = *(const v4f*)&Whh[i * 4];
  for (int i = tid; i < 16 * 64; i += 256) { h_s[i] = 0.f; c_s[i] = 0.f; }
  if (tid < 16) len_s[tid] = lengths[b0 + tid];
  __syncthreads();

  for (int tt = 0; tt < T_; tt++) {
    const int t = dir ? (T_ - 1 - tt) : tt;

    // prefetch gate rows (independent of h): latency hides under WMMA
    float gpre[16];
#pragma unroll
    for (int u = 0; u < 4; u++) {
      int idx = tid + u * 256;
      int bi = idx >> 6, j = idx & 63;
      const float* gr = G + ((size_t)(b0 + bi) * T_ + t) * G4_ + dirOff;
      gpre[u * 4 + 0] = gr[j];
      gpre[u * 4 + 1] = gr[64 + j];
      gpre[u * 4 + 2] = gr[128 + j];
      gpre[u * 4 + 3] = gr[192 + j];
    }

    // S = h @ Whh^T ; wave wv owns gate-tiles 2wv, 2wv+1
    v8f a00 = {}, a01 = {}, a10 = {}, a11 = {};
#pragma unroll
    for (int ks = 0; ks < 16; ks += 2) {
      int kk = ks * 4;
      v2f aE = ld_frag(h_s, 0, 64, kk, lane);
      v2f aO = ld_frag(h_s, 0, 64, kk + 4, lane);
      v2f b0E = ld_frag(whh_s, (2 * wv) * 16, 64, kk, lane);
      a00 = wmma_f32(aE, b0E, a00);
      v2f b1E = ld_frag(whh_s, (2 * wv + 1) * 16, 64, kk, lane);
      a10 = wmma_f32(aE, b1E, a10);
      v2f b0O = ld_frag(whh_s, (2 * wv) * 16, 64, kk + 4, lane);
      a01 = wmma_f32(aO, b0O, a01);
      v2f b1O = ld_frag(whh_s, (2 * wv + 1) * 16, 64, kk + 4, lane);
      a11 = wmma_f32(aO, b1O, a11);
    }
    v8f acc0 = a00 + a01;
    v8f acc1 = a10 + a11;
    {
      const int r0 = ((lane >> 4) << 3);
      const int cl = lane & 15;
#pragma unroll
      for (int i = 0; i < 8; i++) {
        g_s[(r0 + i) * 256 + (2 * wv) * 16 + cl] = acc0[i];
        g_s[(r0 + i) * 256 + (2 * wv + 1) * 16 + cl] = acc1[i];
      }
    }
    __syncthreads();

    // gate math: 16 batch rows x 64 hidden, 4 items/thread
#pragma unroll
    for (int u = 0; u < 4; u++) {
      int idx = tid + u * 256;      // == bi*64 + j
      int bi = idx >> 6, j = idx & 63;
      size_t rowN = (size_t)(b0 + bi) * T_ + t;
      float iv = g_s[bi * 256 + j] + gpre[u * 4 + 0];
      float fv = g_s[bi * 256 + 64 + j] + gpre[u * 4 + 1];
      float gv = g_s[bi * 256 + 128 + j] + gpre[u * 4 + 2];
      float ov = g_s[bi * 256 + 192 + j] + gpre[u * 4 + 3];
      float si = 1.f / (1.f + expf(-iv));
      float sf = 1.f / (1.f + expf(-fv));
      float so = 1.f / (1.f + expf(-ov));
      float tg = tanhf(gv);
      float cOld = c_s[idx];
      float cNew = sf * cOld + si * tg;
      float hNew = so * tanhf(cNew);
      bool valid = t < len_s[bi];
      float h2 = valid ? hNew : h_s[idx];
      float c2 = valid ? cNew : cOld;
      h_s[idx] = h2;
      c_s[idx] = c2;
      float outv = (maskOutput && !valid) ? 0.f : h2;
      Hout[rowN * 128 + outOff + j] = outv;
    }
    __syncthreads();
  }
}

// ---------------------------------------------------------------------------
// K4: attention scores + softmax + pooled feature + both MLP heads.
// ---------------------------------------------------------------------------
__global__ __launch_bounds__(256) void attn_heads(
    const float* __restrict__ hid,  // NROWS x 64, already tanh'd
    const float* __restrict__ H1,   // NROWS x 128, padded steps zeroed
    const int* __restrict__ lengths, const float* a2w, const float* a2b,
    const float* xw1, const float* xb1, const float* xbg, const float* xbb,
    const float* xbm, const float* xbv, const float* xw2, const float* xb2,
    const float* xw3, const float* xb3, const float* tw1, const float* tb1,
    const float* tbg, const float* tbb, const float* tbm, const float* tbv,
    const float* tw2, const float* tb2, const float* tw3, const float* tb3,
    float* __restrict__ out) {
  __shared__ float sc[256];
  __shared__ float red[256];
  __shared__ float af[128];
  __shared__ float z1[128];
  __shared__ float z2[64];
  const int b = blockIdx.x;
  const int tid = threadIdx.x;
  const int len = lengths[b];

  {
    const float* hr = hid + ((size_t)b * T_ + tid) * 64;
    float s = a2b[0];
    for (int j = 0; j < 64; j++) s += a2w[j] * hr[j];
    sc[tid] = (tid < len) ? s : -1e9f;
  }
  __syncthreads();
  red[tid] = sc[tid];
  __syncthreads();
  for (int off = 128; off > 0; off >>= 1) {
    if (tid < off) red[tid] = fmaxf(red[tid], red[tid + off]);
    __syncthreads();
  }
  const float mx = red[0];
  __syncthreads();
  float e = expf(sc[tid] - mx);
  red[tid] = e;
  __syncthreads();
  for (int off = 128; off > 0; off >>= 1) {
    if (tid < off) red[tid] += red[tid + off];
    __syncthreads();
  }
  const float inv = 1.f / red[0];
  __syncthreads();
  sc[tid] = e * inv;                 // softmax weight
  __syncthreads();

  if (tid < 128) {
    float a = 0.f;
    for (int t = 0; t < T_; t++)
      a += sc[t] * H1[((size_t)b * T_ + t) * 128 + tid];
    af[tid] = a;
  }
  __syncthreads();

  // fc_xyz: relu(lin1 128->128), bn, relu(lin2 128->64), lin3 64->3
  if (tid < 128) {
    float s = xb1[tid];
    const float* wr = xw1 + tid * 128;
    for (int k = 0; k < 128; k++) s += wr[k] * af[k];
    s = fmaxf(s, 0.f);
    z1[tid] = (s - xbm[tid]) * (xbg[tid] * rsqrtf(xbv[tid] + EPS_)) + xbb[tid];
  }
  __syncthreads();
  if (tid < 64) {
    float s = xb2[tid];
    const float* wr = xw2 + tid * 128;
    for (int k = 0; k < 128; k++) s += wr[k] * z1[k];
    z2[tid] = fmaxf(s, 0.f);
  }
  __syncthreads();
  if (tid < 3) {
    float s = xb3[tid];
    const float* wr = xw3 + tid * 64;
    for (int k = 0; k < 64; k++) s += wr[k] * z2[k];
    out[b * 3 + tid] = s;
  }
  __syncthreads();

  // fc_time: relu(lin1 128->64), bn, relu(lin2 64->32), lin3 32->1
  if (tid < 64) {
    float s = tb1[tid];
    const float* wr = tw1 + tid * 128;
    for (int k = 0; k < 128; k++) s += wr[k] * af[k];
    s = fmaxf(s, 0.f);
    z1[tid] = (s - tbm[tid]) * (tbg[tid] * rsqrtf(tbv[tid] + EPS_)) + tbb[tid];
  }
  __syncthreads();
  if (tid < 32) {
    float s = tb2[tid];
    const float* wr = tw2 + tid * 64;
    for (int k = 0; k < 64; k++) s += wr[k] * z1[k];
    z2[tid] = fmaxf(s, 0.f);
  }
  __syncthreads();
  if (tid == 0) {
    float s = tb3[0];
    for (int k = 0; k < 32; k++) s += tw3[k] * z2[k];
    out[B_ * 3 + b] = s;
  }
}

// ---------------------------------------------------------------------------
// Host launcher. Param indices follow setup_inputs() recursive dict order:
//  0 x, 1 lengths,
//  block1: 2 c1w 3 c1b 4-7 bn1(g,b,m,v) 8 c2w 9 c2b 10-13 bn2 14 scw 15 scb
//          16-19 sc_bn ; block2: 20..37 same layout
//  lstm: l0_f 38-41 (Wih,Whh,bih,bhh), l0_b 42-45, l1_f 46-49, l1_b 50-53
//  attn: 54 l1w 55 l1b 56 l2w 57 l2b
//  fc_xyz: 58 w1 59 b1 60-63 bn 64 w2 65 b2 66 w3 67 b3
//  fc_time: 68 w1 69 b1 70-73 bn 74 w2 75 b2 76 w3 77 b3
// Workspace (floats): X0 22,020,096 | G0 8,388,608 | H0 2,097,152 |
//                     H1 2,097,152 | AH 1,048,576 ; G1 aliases X0. ~143 MB.
// ---------------------------------------------------------------------------
extern "C" void kernel_launch(void* const* d_in, const int* in_sizes, int n_in,
                              void* d_out, int out_size, void* d_ws,
                              size_t ws_size, hipStream_t stream) {
  (void)in_sizes; (void)n_in; (void)out_size; (void)ws_size;
  auto F = [&](int i) { return (const float*)d_in[i]; };
  const float* x = F(0);
  const int* lengths = (const int*)d_in[1];

  ConvParams p1 = {F(2),  F(3),  F(4),  F(5),  F(6),  F(7),
                   F(8),  F(9),  F(10), F(11), F(12), F(13),
                   F(14), F(15), F(16), F(17), F(18), F(19)};
  ConvParams p2 = {F(20), F(21), F(22), F(23), F(24), F(25),
                   F(26), F(27), F(28), F(29), F(30), F(31),
                   F(32), F(33), F(34), F(35), F(36), F(37)};

  float* ws = (float*)d_ws;
  float* X0 = ws;                         // 16384 x 1344
  float* G0 = X0 + (size_t)NROWS_ * FEAT_;// 16384 x 512
  float* H0 = G0 + (size_t)NROWS_ * G4_;  // 16384 x 128
  float* H1 = H0 + (size_t)NROWS_ * 128;  // 16384 x 128
  float* AH = H1 + (size_t)NROWS_ * 128;  // 16384 x 64
  float* G1 = X0;                         // alias: conv_seq dead after G0 GEMM

  conv_blocks<<<dim3(NROWS_ / 4), dim3(256), 0, stream>>>(x, p1, p2, X0);

  wmma_gemm_xwT<<<dim3(G4_ / 64, NROWS_ / 128), dim3(256), 0, stream>>>(
      X0, F(38), F(42), 256, F(40), F(41), F(44), F(45), G0, NROWS_, G4_,
      FEAT_, 0);

  lstm_layer<<<dim3(8), dim3(256), 0, stream>>>(G0, F(39), F(43), lengths, H0,
                                                0);

  wmma_gemm_xwT<<<dim3(G4_ / 64, NROWS_ / 128), dim3(256), 0, stream>>>(
      H0, F(46), F(50), 256, F(48), F(49), F(52), F(53), G1, NROWS_, G4_, 128,
      0);

  lstm_layer<<<dim3(8), dim3(256), 0, stream>>>(G1, F(47), F(51), lengths, H1,
                                                1);

  wmma_gemm_xwT<<<dim3(1, NROWS_ / 128), dim3(256), 0, stream>>>(
      H1, F(54), F(54), 64, F(55), nullptr, nullptr, nullptr, AH, NROWS_, 64,
      128, 1);

  attn_heads<<<dim3(B_), dim3(256), 0, stream>>>(
      AH, H1, lengths, F(56), F(57), F(58), F(59), F(60), F(61), F(62), F(63),
      F(64), F(65), F(66), F(67), F(68), F(69), F(70), F(71), F(72), F(73),
      F(74), F(75), F(76), F(77), (float*)d_out);
}